// DiceLoss_24386824307351
// MI455X (gfx1250) — compile-verified
//
#include <hip/hip_runtime.h>
#include <hip/hip_bf16.h>

typedef __attribute__((ext_vector_type(2))) float v2f;
typedef __attribute__((ext_vector_type(8))) float v8f;

#define ROW_N       (512 * 512)     // elements per batch row = 262144
#define NBATCH      64
#define BLK_PER_ROW 8
#define THREADS     256
#define CHUNK       (ROW_N / BLK_PER_ROW)          // 32768 elems per block
#define V4_PER_THR  (CHUNK / (THREADS * 4))        // 32 float4 per thread
#define SMOOTH      1.0f

// ---------------------------------------------------------------------------
// Stage 1: per-(row, block) partial sums of p, t, p*t. Pure bandwidth kernel:
// float4 loads -> global_load_b128, fma accumulation, wave32 shuffle reduce,
// LDS cross-wave reduce, deterministic partials to workspace (no atomics).
// ---------------------------------------------------------------------------
__global__ void __launch_bounds__(THREADS)
dice_partial_kernel(const float* __restrict__ pred,
                    const float* __restrict__ targ,
                    float* __restrict__ part /* [NBATCH*BLK_PER_ROW*3] */) {
    const int r   = blockIdx.y;
    const int bx  = blockIdx.x;
    const int tid = threadIdx.x;

    const size_t base = (size_t)r * ROW_N + (size_t)bx * CHUNK;
    const float4* __restrict__ p4 = (const float4*)(pred + base);
    const float4* __restrict__ t4 = (const float4*)(targ + base);

    float ps = 0.0f, ts = 0.0f, is = 0.0f;
#pragma unroll 4
    for (int i = 0; i < V4_PER_THR; ++i) {
        const float4 p = p4[i * THREADS + tid];   // coalesced 128-bit loads
        const float4 t = t4[i * THREADS + tid];
        ps += (p.x + p.y) + (p.z + p.w);
        ts += (t.x + t.y) + (t.z + t.w);
        is = fmaf(p.x, t.x, fmaf(p.y, t.y, fmaf(p.z, t.z, fmaf(p.w, t.w, is))));
    }

    // wave32 butterfly reduction
#pragma unroll
    for (int off = 16; off > 0; off >>= 1) {
        ps += __shfl_down(ps, off);
        ts += __shfl_down(ts, off);
        is += __shfl_down(is, off);
    }

    __shared__ float red[3][THREADS / 32];
    const int wave = tid >> 5;
    if ((tid & 31) == 0) {
        red[0][wave] = ps;
        red[1][wave] = ts;
        red[2][wave] = is;
    }
    __syncthreads();

    if (tid == 0) {
        float a = 0.0f, b = 0.0f, c = 0.0f;
#pragma unroll
        for (int w = 0; w < THREADS / 32; ++w) {
            a += red[0][w];
            b += red[1][w];
            c += red[2][w];
        }
        float* q = part + (size_t)(r * BLK_PER_ROW + bx) * 3;
        q[0] = a;   // p_sum partial
        q[1] = b;   // t_sum partial
        q[2] = c;   // inter partial
    }
}

// ---------------------------------------------------------------------------
// Stage 2: one wave (32 threads, EXEC all-ones) computes the 64 per-row
// losses and reduces them with V_WMMA_F32_16X16X4_F32 (B = ones => row sums
// of the 16x4 A matrix holding all 64 losses), then a single shfl combines
// the two half-results and lane 0 writes the mean.
// ---------------------------------------------------------------------------
__device__ __forceinline__ float row_loss(const float* __restrict__ part, int r) {
    float ps = 0.0f, ts = 0.0f, is = 0.0f;
#pragma unroll
    for (int b = 0; b < BLK_PER_ROW; ++b) {
        const float* q = part + (size_t)(r * BLK_PER_ROW + b) * 3;
        ps += q[0];
        ts += q[1];
        is += q[2];
    }
    const float dice      = (2.0f * is + SMOOTH) / (ps + ts + SMOOTH);
    const float dice_loss = 1.0f - dice;
    const float zero_loss = ps * (1.0f / (float)ROW_N);
    // exact +0.0 compare matches jnp.where(t_sum == 0, ...); lowers to v_cndmask
    return (ts == 0.0f) ? zero_loss : dice_loss;
}

__global__ void __launch_bounds__(32)
dice_finalize_kernel(const float* __restrict__ part, float* __restrict__ out) {
    const int lane = threadIdx.x;   // 0..31, full wave, EXEC all-ones

    // A-matrix 16x4 f32 layout (ISA 7.12.2):
    //   lane m    (m<16): VGPR0 = A[m][0], VGPR1 = A[m][1]
    //   lane m+16       : VGPR0 = A[m][2], VGPR1 = A[m][3]
    // Pack: A[m][0]=L[m], A[m][1]=L[m+32], A[m][2]=L[m+16], A[m][3]=L[m+48]
    // => rowsum_m = L[m] + L[m+16] + L[m+32] + L[m+48]; sum_m rowsum = total.
    v2f a;
    a.x = row_loss(part, lane);
    a.y = row_loss(part, lane + 32);

    v2f b;                       // B = all-ones 4x16 (layout-independent)
    b.x = 1.0f;
    b.y = 1.0f;

    v8f c = {};
    c = __builtin_amdgcn_wmma_f32_16x16x4_f32(
            /*neg_a=*/false, a, /*neg_b=*/false, b,
            /*c_mod=*/(short)0, c, /*reuse_a=*/false, /*reuse_b=*/false);

    // C/D layout: lane L, VGPR v holds D[v + 8*(L>=16)][L%16] = rowsum.
    // lane 0 covers M=0..7, lane 16 covers M=8..15.
    float s = ((c[0] + c[1]) + (c[2] + c[3])) + ((c[4] + c[5]) + (c[6] + c[7]));
    float total = s + __shfl(s, 16);   // lane 0: M0..7 sums + M8..15 sums

    if (lane == 0) {
        out[0] = total * (1.0f / (float)NBATCH);
    }
}

// ---------------------------------------------------------------------------
extern "C" void kernel_launch(void* const* d_in, const int* in_sizes, int n_in,
                              void* d_out, int out_size, void* d_ws, size_t ws_size,
                              hipStream_t stream) {
    const float* pred = (const float*)d_in[0];
    const float* targ = (const float*)d_in[1];
    float* out  = (float*)d_out;
    float* part = (float*)d_ws;   // 64 * 8 * 3 floats = 6144 bytes

    dim3 grid(BLK_PER_ROW, NBATCH);
    dice_partial_kernel<<<grid, THREADS, 0, stream>>>(pred, targ, part);
    dice_finalize_kernel<<<1, 32, 0, stream>>>(part, out);
}